// Model_HY_30090540875814
// MI455X (gfx1250) — compile-verified
//
#include <hip/hip_runtime.h>

#define LAT    32
#define EPSV   1e-5f

typedef float v2f __attribute__((ext_vector_type(2)));
typedef float v8f __attribute__((ext_vector_type(8)));

__device__ __forceinline__ v8f wmma_f32(v2f a, v2f b, v8f c) {
    // V_WMMA_F32_16X16X4_F32 : D = A(16x4) x B(4x16) + C(16x16), fp32
    return __builtin_amdgcn_wmma_f32_16x16x4_f32(
        /*neg_a=*/false, a, /*neg_b=*/false, b,
        /*c_mod=*/(short)0, c, /*reuse_a=*/false, /*reuse_b=*/false);
}

// ---------------------------------------------------------------- utilities
__global__ void k_zero(float* __restrict__ p, long n) {
    long i = (long)blockIdx.x * blockDim.x + threadIdx.x;
    long stride = (long)gridDim.x * blockDim.x;
    for (; i < n; i += stride) p[i] = 0.0f;
}

// ------------------------------------------------- embedding + column sums
__global__ void k_embed_stats(const int* __restrict__ x, const float* __restrict__ emb,
                              float* __restrict__ h, float* __restrict__ colsum, int n) {
    __shared__ float sdata[256];
    int f  = threadIdx.x & 31;
    int rl = threadIdx.x >> 5;                       // 0..7 row-lane within block
    float part = 0.0f;
    for (int i = blockIdx.x * 8 + rl; i < n; i += gridDim.x * 8) {
        float v = emb[(long)x[i] * LAT + f];
        h[(long)i * LAT + f] = v;
        part += v;
    }
    sdata[threadIdx.x] = part;
    __syncthreads();
    for (int s = 128; s >= 32; s >>= 1) {
        if (threadIdx.x < s) sdata[threadIdx.x] += sdata[threadIdx.x + s];
        __syncthreads();
    }
    if (threadIdx.x < 32) atomicAdd(&colsum[f], sdata[threadIdx.x]);
}

// ------------------------------------- center in place + column sum-squares
__global__ void k_center(float* __restrict__ h, const float* __restrict__ colsum,
                         const float* __restrict__ mscale, float* __restrict__ sumsq, int n) {
    __shared__ float sdata[256];
    int f  = threadIdx.x & 31;
    int rl = threadIdx.x >> 5;
    float mean = colsum[f] / (float)n;
    float ms   = mscale[f];
    float part = 0.0f;
    for (int i = blockIdx.x * 8 + rl; i < n; i += gridDim.x * 8) {
        float v = h[(long)i * LAT + f] - ms * mean;
        h[(long)i * LAT + f] = v;
        part += v * v;
    }
    sdata[threadIdx.x] = part;
    __syncthreads();
    for (int s = 128; s >= 32; s >>= 1) {
        if (threadIdx.x < s) sdata[threadIdx.x] += sdata[threadIdx.x + s];
        __syncthreads();
    }
    if (threadIdx.x < 32) atomicAdd(&sumsq[f], sdata[threadIdx.x]);
}

// ------------------------------------------------------------ normalization
__global__ void k_norm(float* __restrict__ h, const float* __restrict__ sumsq,
                       const float* __restrict__ w, const float* __restrict__ b, int n) {
    long i = (long)blockIdx.x * blockDim.x + threadIdx.x;
    if (i >= (long)n * LAT) return;
    int f = (int)(i & 31);
    float var = sumsq[f] / (float)n;
    h[i] = w[f] * h[i] * rsqrtf(var + EPSV) + b[f];
}

// ------------------------------------------- edge scatter (mean-agg pieces)
// 8 lanes per edge, 4 features per lane -> spreads f32 atomics across L2 banks
__global__ void k_scatter(const int* __restrict__ src, const int* __restrict__ dst,
                          const float* __restrict__ h, float* __restrict__ agg,
                          float* __restrict__ cnt, int ne) {
    int t = blockIdx.x * blockDim.x + threadIdx.x;
    int e = t >> 3;
    if (e >= ne) return;
    int sub = t & 7;
    int s = src[e], d = dst[e];
    const float* hs = h + (long)s * LAT + sub * 4;
    float*       ad = agg + (long)d * LAT + sub * 4;
    atomicAdd(ad + 0, hs[0]);
    atomicAdd(ad + 1, hs[1]);
    atomicAdd(ad + 2, hs[2]);
    atomicAdd(ad + 3, hs[3]);
    if (sub == 0) atomicAdd(&cnt[d], 1.0f);
}

// -------------------------------------------------------- SAGE dense + WMMA
// out[r,:] = relu( (agg[r,:]/max(cnt,1)) @ wl.T + bl + x[r,:] @ wr.T )  (+= if accum)
// One wave computes a 16-row x 32-col tile via chained V_WMMA_F32_16X16X4_F32.
__global__ __launch_bounds__(256) void k_sage_wmma(
    const float* __restrict__ x, const float* __restrict__ agg,
    const float* __restrict__ cnt, const float* __restrict__ wl,
    const float* __restrict__ bl, const float* __restrict__ wr,
    float* __restrict__ out, int n, int accum) {
    int wave = threadIdx.x >> 5;
    int lane = threadIdx.x & 31;
    int half = lane >> 4;                 // ISA A/B layout: lanes 16-31 carry K+2
    int lr   = lane & 15;
    int r0   = (blockIdx.x * 8 + wave) * 16;

    int rowA = r0 + lr;                   // A-fragment row for this lane (M = lane%16)
    int rowC = rowA < n ? rowA : (n - 1); // clamp: EXEC must stay all-1s for WMMA
    float inv = 1.0f / fmaxf(cnt[rowC], 1.0f);
    const float* ar = agg + (long)rowC * LAT;
    const float* xr = x   + (long)rowC * LAT;
    int col0 = lr;                        // B-fragment / output column (N = lane%16)
    int col1 = 16 + lr;

    v8f acc0 = {};                        // cols 0..15
    v8f acc1 = {};                        // cols 16..31

    #pragma unroll
    for (int k = 0; k < LAT; k += 4) {
        int ko = k + 2 * half;
        v2f aAgg, aX, b;
        aAgg.x = ar[ko] * inv;  aAgg.y = ar[ko + 1] * inv;
        aX.x   = xr[ko];        aX.y   = xr[ko + 1];
        // B[kk][c] = W[c][kk]  (matmul is  X @ W.T)
        b.x = wl[col0 * LAT + ko]; b.y = wl[col0 * LAT + ko + 1];
        acc0 = wmma_f32(aAgg, b, acc0);
        b.x = wr[col0 * LAT + ko]; b.y = wr[col0 * LAT + ko + 1];
        acc0 = wmma_f32(aX, b, acc0);
        b.x = wl[col1 * LAT + ko]; b.y = wl[col1 * LAT + ko + 1];
        acc1 = wmma_f32(aAgg, b, acc1);
        b.x = wr[col1 * LAT + ko]; b.y = wr[col1 * LAT + ko + 1];
        acc1 = wmma_f32(aX, b, acc1);
    }

    float b0 = bl[col0], b1 = bl[col1];
    #pragma unroll
    for (int j = 0; j < 8; j++) {
        int row = r0 + j + 8 * half;      // C/D layout: VGPR j -> M=j (lanes 0-15), M=j+8 (16-31)
        if (row < n) {
            float v0 = fmaxf(acc0[j] + b0, 0.0f);
            float v1 = fmaxf(acc1[j] + b1, 0.0f);
            float* o = out + (long)row * LAT;
            if (accum) { o[col0] += v0; o[col1] += v1; }
            else       { o[col0]  = v0; o[col1]  = v1; }
        }
    }
}

// ---------------------------------------------- pair features h[p0] * h[p1]
__global__ void k_pair(const float* __restrict__ h, const int* __restrict__ pos1,
                       float* __restrict__ p, int n2) {
    int i = blockIdx.x * blockDim.x + threadIdx.x;
    int r = i >> 5, f = i & 31;
    if (r >= n2) return;
    int a = pos1[2 * r], b = pos1[2 * r + 1];
    p[(long)r * LAT + f] = h[(long)a * LAT + f] * h[(long)b * LAT + f];
}

// --------------------- final: gather pos2, even*odd rows, dot with pred_w
__global__ void k_final(const float* __restrict__ h, const int* __restrict__ pos2,
                        const float* __restrict__ pw, const float* __restrict__ pb,
                        float* __restrict__ out, int m2) {
    int j = blockIdx.x * blockDim.x + threadIdx.x;
    if (j >= m2) return;
    int a = pos2[2 * j], b = pos2[2 * j + 1];
    const float* ha = h + (long)a * LAT;
    const float* hb = h + (long)b * LAT;
    float s = 0.0f;
    #pragma unroll
    for (int k = 0; k < LAT; k++) s += ha[k] * hb[k] * pw[k];
    out[j] = s + pb[0];
}

// ---------------------------------------------------------------------------
extern "C" void kernel_launch(void* const* d_in, const int* in_sizes, int n_in,
                              void* d_out, int out_size, void* d_ws, size_t ws_size,
                              hipStream_t stream) {
    const int*   x      = (const int*)d_in[0];
    const int*   edge1  = (const int*)d_in[2];
    const int*   edge2  = (const int*)d_in[3];
    const int*   edge2r = (const int*)d_in[4];
    const int*   pos1   = (const int*)d_in[5];
    const int*   pos2   = (const int*)d_in[6];
    const float* emb    = (const float*)d_in[7];
    const float* gw     = (const float*)d_in[8];
    const float* gb     = (const float*)d_in[9];
    const float* gms    = (const float*)d_in[10];
    const float* c1_wl  = (const float*)d_in[11];
    const float* c1_bl  = (const float*)d_in[12];
    const float* c1_wr  = (const float*)d_in[13];
    const float* c2_wl  = (const float*)d_in[14];
    const float* c2_bl  = (const float*)d_in[15];
    const float* c2_wr  = (const float*)d_in[16];
    const float* c2r_wl = (const float*)d_in[17];
    const float* c2r_bl = (const float*)d_in[18];
    const float* c2r_wr = (const float*)d_in[19];
    const float* pred_w = (const float*)d_in[20];
    const float* pred_b = (const float*)d_in[21];
    float* outp = (float*)d_out;

    const int n  = in_sizes[0];          // 100000
    const int e1 = in_sizes[2] / 2;      // 3200000
    const int e2 = in_sizes[3] / 2;      // 3200000
    const int e2r = in_sizes[4] / 2;     // 3200000
    const int n2 = in_sizes[5] / 2;      // 200000
    const int m  = in_sizes[6];          // 131072
    const int m2 = m / 2;

    // workspace layout (floats)
    float* ws = (float*)d_ws;
    const size_t cap = (size_t)n2 * LAT;
    float* bufA  = ws;
    float* bufB  = bufA + cap;
    float* aggA  = bufB + cap;
    float* aggB  = aggA + cap;
    float* cntA  = aggB + cap;
    float* cntB  = cntA + n2;
    float* stats = cntB + n2;            // [0:32) colsum, [32:64) sumsq

    const int B = 256;
    auto grd = [](long cnt, int b) { return (int)((cnt + b - 1) / b); };

    // ---- embedding + GraphNorm
    k_zero<<<1, 64, 0, stream>>>(stats, 64);
    k_embed_stats<<<512, B, 0, stream>>>(x, emb, bufA, stats, n);
    k_center<<<512, B, 0, stream>>>(bufA, stats, gms, stats + 32, n);
    k_norm<<<grd((long)n * LAT, B), B, 0, stream>>>(bufA, stats + 32, gw, gb, n);

    // ---- graph-1 SAGE layers
    float* cur = bufA;
    float* nxt = bufB;
    for (int i = 0; i < 2; i++) {
        k_zero<<<8192, B, 0, stream>>>(aggA, (long)n * LAT);
        k_zero<<<1024, B, 0, stream>>>(cntA, n);
        k_scatter<<<grd((long)e1 * 8, B), B, 0, stream>>>(edge1, edge1 + e1, cur, aggA, cntA, e1);
        k_sage_wmma<<<grd(n, 128), B, 0, stream>>>(cur, aggA, cntA,
                                                   c1_wl + i * LAT * LAT, c1_bl + i * LAT,
                                                   c1_wr + i * LAT * LAT, nxt, n, 0);
        float* t = cur; cur = nxt; nxt = t;
    }

    // ---- pair features
    k_pair<<<grd((long)n2 * LAT, B), B, 0, stream>>>(cur, pos1, nxt, n2);
    { float* t = cur; cur = nxt; nxt = t; }

    // ---- graph-2 dual SAGE layers:  relu(sage_fwd) + relu(sage_rev)
    for (int i = 0; i < 2; i++) {
        k_zero<<<8192, B, 0, stream>>>(aggA, (long)n2 * LAT);
        k_zero<<<1024, B, 0, stream>>>(cntA, n2);
        k_zero<<<8192, B, 0, stream>>>(aggB, (long)n2 * LAT);
        k_zero<<<1024, B, 0, stream>>>(cntB, n2);
        k_scatter<<<grd((long)e2 * 8, B), B, 0, stream>>>(edge2, edge2 + e2, cur, aggA, cntA, e2);
        k_scatter<<<grd((long)e2r * 8, B), B, 0, stream>>>(edge2r, edge2r + e2r, cur, aggB, cntB, e2r);
        k_sage_wmma<<<grd(n2, 128), B, 0, stream>>>(cur, aggA, cntA,
                                                    c2_wl + i * LAT * LAT, c2_bl + i * LAT,
                                                    c2_wr + i * LAT * LAT, nxt, n2, 0);
        k_sage_wmma<<<grd(n2, 128), B, 0, stream>>>(cur, aggB, cntB,
                                                    c2r_wl + i * LAT * LAT, c2r_bl + i * LAT,
                                                    c2r_wr + i * LAT * LAT, nxt, n2, 1);
        float* t = cur; cur = nxt; nxt = t;
    }

    // ---- final gather + even*odd product + 32->1 linear
    k_final<<<grd(m2, B), B, 0, stream>>>(cur, pos2, pred_w, pred_b, outp, m2);
}